// DietConv2dV2_2499670966535
// MI455X (gfx1250) — compile-verified
//
#include <hip/hip_runtime.h>
#include <hip/hip_bf16.h>

typedef float v2f __attribute__((ext_vector_type(2)));
typedef float v8f __attribute__((ext_vector_type(8)));

#define HH    1024
#define WW    1024
#define CIN   8
#define COUT  8
#define KTOT  72          // CIN * 3 * 3
#define NCR   24          // (c,r) pairs = CIN*3
#define JROW  130         // 128 pixels + 2 halo columns
#define JSTR  26          // LDS dwords per j column (24 cr slots + 2 pad -> even offsets)
#define WTILE 128         // pixels per workgroup (8 waves x 16 pixels)

// Implicit-GEMM conv3x3 (stride 1, pad 1) with V_WMMA_F32_16X16X4_F32.
// K ordering u = s*24 + (c*3+r); LDS laid out so every WMMA step's A/B
// operands are one aligned ds_load_b64 at a compile-time immediate offset.
// Grid: (W/128, H, N). Block: 256 threads (8 wave32).
__global__ __launch_bounds__(256) void conv3x3_wmma_f32(
    const float* __restrict__ x,     // [N][CIN][H][W]
    const float* __restrict__ wgt,   // [COUT][CIN][3][3]
    const float* __restrict__ bias,  // [COUT]
    float* __restrict__ out)         // [N][COUT][H][W]
{
    __shared__ float sx[JROW * JSTR];   // [j][cr] ; j=0 <-> w = w0-1
    __shared__ float sw[COUT * KTOT];   // [o][u]  ; u = s*24 + c*3 + r

    const int tid  = threadIdx.x;
    const int wblk = blockIdx.x;       // 0..7
    const int h    = blockIdx.y;       // 0..1023
    const int n    = blockIdx.z;       // 0..15
    const int w0   = wblk * WTILE;

    // ---- Stage input patch, j-major, zero-padded halo ----
    // idx = cr*JROW + j: consecutive threads read consecutive global w.
    for (int idx = tid; idx < NCR * JROW; idx += 256) {
        int cr = idx / JROW;
        int j  = idx - cr * JROW;
        int c  = cr / 3;
        int r  = cr - 3 * c;
        int wc = w0 - 1 + j;
        int hc = h - 1 + r;
        float v = 0.0f;
        if ((unsigned)wc < WW && (unsigned)hc < HH)
            v = x[(((size_t)n * CIN + c) * HH + hc) * WW + wc];
        sx[j * JSTR + cr] = v;
    }
    // ---- Stage weights permuted into u-order: sw[o*72+u] = wgt[o][c][r][s] ----
    for (int idx = tid; idx < COUT * KTOT; idx += 256) {
        int o  = idx / KTOT;
        int u  = idx - o * KTOT;
        int s  = u / NCR;
        int cr = u - s * NCR;
        sw[idx] = wgt[o * KTOT + (cr / 3) * 9 + (cr % 3) * 3 + s];
    }
    __syncthreads();

    const int lane = tid & 31;
    const int wave = tid >> 5;
    const int hi   = lane >> 4;     // half-wave: K offset +2
    const int lo16 = lane & 15;
    const int p0   = wave * 16;     // wave's pixel base inside the 128-strip
    const int m    = lo16;          // A-matrix row (pixel offset)
    const int o    = lo16 & 7;      // clamped output channel

    // Per-lane base pointers (all step offsets are compile-time immediates).
    const float* aBase = &sx[(p0 + m) * JSTR + 2 * hi];   // 8B-aligned
    const float* bBase = &sw[o * KTOT + 2 * hi];          // 8B-aligned

    // Accumulator preloaded with bias (this lane's column is channel o).
    v8f acc;
    const float bv = bias[o];
    #pragma unroll
    for (int r = 0; r < 8; ++r) acc[r] = bv;

    // 18 chained K=4 WMMA steps. A offset: 4t + 2*(t/6) dwords (always even).
    #pragma unroll
    for (int t = 0; t < 18; ++t) {
        const int offA = 4 * t + 2 * (t / 6);   // u + 2*(u/24) at u=4t
        v2f a = *(const v2f*)(aBase + offA);    // ds_load_b64, imm offset
        v2f b = *(const v2f*)(bBase + 4 * t);   // ds_load_b64, imm offset
        acc = __builtin_amdgcn_wmma_f32_16x16x4_f32(
            /*neg_a=*/false, a, /*neg_b=*/false, b,
            /*c_mod=*/(short)0, acc, /*reuse_a=*/false, /*reuse_b=*/false);
    }

    // ---- Store: C layout VGPR r -> pixel (r + 8*hi), column = channel ----
    if (lo16 < COUT) {
        const int wbase = w0 + p0 + hi * 8;
        float* op = out + (((size_t)n * COUT + o) * HH + h) * WW + wbase;
        #pragma unroll
        for (int r = 0; r < 8; ++r) op[r] = acc[r];
    }
}

extern "C" void kernel_launch(void* const* d_in, const int* in_sizes, int n_in,
                              void* d_out, int out_size, void* d_ws, size_t ws_size,
                              hipStream_t stream) {
    const float* x    = (const float*)d_in[0];
    const float* wgt  = (const float*)d_in[1];
    const float* bias = (const float*)d_in[2];
    float* out        = (float*)d_out;

    dim3 grid(WW / WTILE, HH, 16);
    dim3 block(256);
    conv3x3_wmma_f32<<<grid, block, 0, stream>>>(x, wgt, bias, out);
}